// GCN_61967788146721
// MI455X (gfx1250) — compile-verified
//
#include <hip/hip_runtime.h>

#define N_NODES 100000
#define N_EDGES 1600000
#define IN_C 26
#define HID_C 128
#define OUT_C 64
#define AS_STRIDE 132   // 128 + 4: keeps column reads on distinct LDS banks

typedef __attribute__((ext_vector_type(2))) float v2f;
typedef __attribute__((ext_vector_type(8))) float v8f;

__device__ __constant__ float WT[IN_C] = {
    0.7f,0.9f,0.7f,0.9f,0.3f,0.7f,0.3f,0.9f,0.3f,0.3f,0.9f,0.7f,0.1f,
    0.9f,0.5f,0.9f,0.5f,0.5f,0.1f,0.3f,0.7f,0.9f,0.9f,0.9f,0.9f,0.9f};

__device__ __forceinline__ void atomic_add_f32(float* p, float v) {
    __hip_atomic_fetch_add(p, v, __ATOMIC_RELAXED, __HIP_MEMORY_SCOPE_AGENT);
}

// ---------------- degree / norm ----------------
__global__ void gcn_degree(const int* __restrict__ src, const int* __restrict__ dst,
                           float* __restrict__ degs, float* __restrict__ degd, int n_edges) {
    int e = blockIdx.x * blockDim.x + threadIdx.x;
    if (e >= n_edges) return;
    atomic_add_f32(&degs[src[e]], 1.0f);
    atomic_add_f32(&degd[dst[e]], 1.0f);
}

__global__ void gcn_norm(float* __restrict__ d, int n) {
    int i = blockIdx.x * blockDim.x + threadIdx.x;
    if (i < n) d[i] = rsqrtf(fmaxf(d[i], 1.0f));   // clip(deg,1)^-0.5
}

// ---------------- node weights (argmax + table) ----------------
__global__ void gcn_node_weights(const float* __restrict__ h, float* __restrict__ w, int n) {
    int i = blockIdx.x * blockDim.x + threadIdx.x;
    if (i >= n) return;
    const float* row = h + i * IN_C;
    float best = row[0]; int bi = 0;
    #pragma unroll
    for (int c = 1; c < IN_C; ++c) {
        float v = row[c];
        if (v > best) { best = v; bi = c; }        // first-max wins, like jnp.argmax
    }
    w[i] = WT[bi];
}

// ---------------- WMMA f32 GEMM with fused BN/ReLU/row-scale on A ----------------
// C[m, outc] = (f(A[m,:K])) @ W[K, outc] (+ bias), where
// f(v) = relu_in ? max(v*colscale+colshift, 0) : v, then optionally * rowscale[m].
// blockDim.x = 32 * (outc/16); each wave computes one 16x16 output tile.
// Both A (16 x Kpad) and B (Kpad x outc, stored transposed) are staged in LDS,
// zero-padded, so the inner loop is branch-free: 2x ds_load_b64 + v_wmma.
__global__ void gcn_wmma_gemm(const float* __restrict__ A, int lda, int K,
                              const float* __restrict__ W, int outc,
                              const float* __restrict__ rowscale,
                              const float* __restrict__ colscale,
                              const float* __restrict__ colshift,
                              int relu_in,
                              const float* __restrict__ bias,
                              float* __restrict__ C, int n_rows) {
    __shared__ float As[16 * AS_STRIDE];          // As[r][k]
    __shared__ float Bs[HID_C * AS_STRIDE];       // Bs[col][k]  (transposed)
    const int Kpad = (K + 3) & ~3;
    const int m0 = blockIdx.x * 16;

    // Stage A tile with per-input fusion (BN scale/shift, ReLU, row norm).
    for (int idx = threadIdx.x; idx < 16 * Kpad; idx += blockDim.x) {
        int r = idx / Kpad;
        int k = idx - r * Kpad;
        int row = m0 + r;
        float v = 0.0f;
        if (row < n_rows && k < K) {
            v = A[row * lda + k];
            if (colscale) v = v * colscale[k] + colshift[k];
            if (relu_in)  v = fmaxf(v, 0.0f);
            if (rowscale) v *= rowscale[row];
        }
        As[r * AS_STRIDE + k] = v;
    }

    // Stage B transposed, float4-vectorized along columns (outc % 4 == 0).
    const int nvec = (Kpad * outc) >> 2;
    for (int idx = threadIdx.x; idx < nvec; idx += blockDim.x) {
        int base = idx << 2;                      // flat k*outc + c, c % 4 == 0
        int k = base / outc;
        int c = base - k * outc;
        float4 v = make_float4(0.f, 0.f, 0.f, 0.f);
        if (k < K) v = *(const float4*)(W + k * outc + c);
        Bs[(c + 0) * AS_STRIDE + k] = v.x;
        Bs[(c + 1) * AS_STRIDE + k] = v.y;
        Bs[(c + 2) * AS_STRIDE + k] = v.z;
        Bs[(c + 3) * AS_STRIDE + k] = v.w;
    }
    __syncthreads();

    const int lane = threadIdx.x & 31;
    const int wave = threadIdx.x >> 5;
    const int col0 = wave << 4;
    const int am   = lane & 15;           // M index of A frag
    const int koff = (lane >> 4) << 1;    // lanes 0-15: K=0,1 ; lanes 16-31: K=2,3
    const int col  = col0 + (lane & 15);  // N index of B frag

    const float* Ap = &As[am * AS_STRIDE + koff];
    const float* Bp = &Bs[col * AS_STRIDE + koff];

    v8f acc = {0.f,0.f,0.f,0.f,0.f,0.f,0.f,0.f};
    #pragma unroll 4
    for (int kk = 0; kk < Kpad; kk += 4) {
        v2f a, b;
        a.x = Ap[kk];
        a.y = Ap[kk + 1];
        b.x = Bp[kk];
        b.y = Bp[kk + 1];
        acc = __builtin_amdgcn_wmma_f32_16x16x4_f32(
                  false, a, false, b, (short)0, acc, false, false);
    }

    // C/D layout: VGPR i -> M = i + 8*(lane>=16), N = lane&15
    const int mb = (lane >> 4) << 3;
    const float bv = bias ? bias[col] : 0.0f;
    #pragma unroll
    for (int i = 0; i < 8; ++i) {
        int m = m0 + mb + i;
        if (m < n_rows) C[m * outc + col] = acc[i] + bv;
    }
}

// ---------------- edge scatter: agg[dst] += y[src] (128 ch, float4 granules) ----------------
__global__ void gcn_scatter(const float* __restrict__ y, const int* __restrict__ src,
                            const int* __restrict__ dst, float* __restrict__ agg, int n_edges) {
    long long idx = (long long)blockIdx.x * blockDim.x + threadIdx.x;
    long long total = (long long)n_edges * (HID_C / 4);
    if (idx >= total) return;
    int e  = (int)(idx >> 5);             // 32 float4-groups per edge (128 ch)
    int cg = ((int)idx & 31) << 2;
    int s = src[e], d = dst[e];
    const float4 v = *(const float4*)(y + s * HID_C + cg);
    float* p = agg + d * HID_C + cg;
    atomic_add_f32(p + 0, v.x);
    atomic_add_f32(p + 1, v.y);
    atomic_add_f32(p + 2, v.z);
    atomic_add_f32(p + 3, v.w);
}

// ---------------- fused (agg*norm_dst + b) in place + BN sum/sumsq ----------------
__global__ void gcn_bn_stats(float* __restrict__ x, const float* __restrict__ norm_dst,
                             const float* __restrict__ bias,
                             float* __restrict__ bsum, float* __restrict__ bsq, int n_rows) {
    int tid = blockIdx.x * blockDim.x + threadIdx.x;
    int c  = tid & (HID_C - 1);
    int r0 = tid >> 7;
    int rs = (gridDim.x * blockDim.x) >> 7;
    float s = 0.f, s2 = 0.f;
    float bc = bias[c];
    for (int r = r0; r < n_rows; r += rs) {
        float t = x[r * HID_C + c] * norm_dst[r] + bc;
        x[r * HID_C + c] = t;
        s += t; s2 += t * t;
    }
    atomic_add_f32(&bsum[c], s);
    atomic_add_f32(&bsq[c], s2);
}

__global__ void gcn_bn_finalize(const float* __restrict__ bsum, const float* __restrict__ bsq,
                                const float* __restrict__ gamma, const float* __restrict__ beta,
                                float* __restrict__ scale, float* __restrict__ shift, float inv_n) {
    int c = threadIdx.x;
    float mu  = bsum[c] * inv_n;
    float var = bsq[c] * inv_n - mu * mu;
    float sc  = gamma[c] * rsqrtf(var + 1e-5f);
    scale[c] = sc;
    shift[c] = beta[c] - mu * sc;
}

// ---------------- orchestration ----------------
extern "C" void kernel_launch(void* const* d_in, const int* in_sizes, int n_in,
                              void* d_out, int out_size, void* d_ws, size_t ws_size,
                              hipStream_t stream) {
    (void)in_sizes; (void)n_in; (void)out_size; (void)ws_size;
    const float* h   = (const float*)d_in[0];
    const int*   src = (const int*)d_in[1];
    const int*   dst = (const int*)d_in[2];
    const float* w0  = (const float*)d_in[3];
    const float* b0  = (const float*)d_in[4];
    const float* w1  = (const float*)d_in[5];
    const float* b1  = (const float*)d_in[6];
    const float* w2  = (const float*)d_in[7];
    const float* b2  = (const float*)d_in[8];
    const float* g0  = (const float*)d_in[9];
    const float* bt0 = (const float*)d_in[10];
    const float* g1  = (const float*)d_in[11];
    const float* bt1 = (const float*)d_in[12];
    const float* g2  = (const float*)d_in[13];
    const float* bt2 = (const float*)d_in[14];
    const float* wfc = (const float*)d_in[15];
    const float* bfc = (const float*)d_in[16];

    float* out   = (float*)d_out;                       // [N, 64]
    float* nodew = out + (size_t)N_NODES * OUT_C;       // [N, 1]

    float* ws = (float*)d_ws;
    float* norm_src = ws;                               // N
    float* norm_dst = ws + N_NODES;                     // N
    float* ybuf     = ws + 2 * N_NODES;                 // N*128
    float* agg      = ybuf + (size_t)N_NODES * HID_C;   // N*128
    float* bsum     = agg + (size_t)N_NODES * HID_C;    // 128
    float* bsq      = bsum + HID_C;                     // 128
    float* bnscale  = bsq + HID_C;                      // 128
    float* bnshift  = bnscale + HID_C;                  // 128

    // degrees -> norms (both buffers contiguous, one transform launch)
    hipMemsetAsync(norm_src, 0, 2 * N_NODES * sizeof(float), stream);
    gcn_degree<<<(N_EDGES + 255) / 256, 256, 0, stream>>>(src, dst, norm_src, norm_dst, N_EDGES);
    gcn_norm<<<(2 * N_NODES + 255) / 256, 256, 0, stream>>>(norm_src, 2 * N_NODES);

    gcn_node_weights<<<(N_NODES + 255) / 256, 256, 0, stream>>>(h, nodew, N_NODES);

    const int gemm_blocks = (N_NODES + 15) / 16;        // 6250 full tiles
    const int scat_blocks = (int)(((long long)N_EDGES * (HID_C / 4) + 255) / 256);

    const float* Ws[3] = {w0, w1, w2};
    const float* Bs[3] = {b0, b1, b2};
    const float* Gs[3] = {g0, g1, g2};
    const float* Bt[3] = {bt0, bt1, bt2};

    for (int l = 0; l < 3; ++l) {
        if (l == 0) {
            gcn_wmma_gemm<<<gemm_blocks, 32 * (HID_C / 16), 0, stream>>>(
                h, IN_C, IN_C, Ws[0], HID_C,
                norm_src, nullptr, nullptr, 0, nullptr, ybuf, N_NODES);
        } else {
            // fused: relu(bn(prev)) * norm_src, staged into LDS, then WMMA
            gcn_wmma_gemm<<<gemm_blocks, 32 * (HID_C / 16), 0, stream>>>(
                agg, HID_C, HID_C, Ws[l], HID_C,
                norm_src, bnscale, bnshift, 1, nullptr, ybuf, N_NODES);
        }
        hipMemsetAsync(agg, 0, (size_t)N_NODES * HID_C * sizeof(float), stream);
        gcn_scatter<<<scat_blocks, 256, 0, stream>>>(ybuf, src, dst, agg, N_EDGES);
        hipMemsetAsync(bsum, 0, 2 * HID_C * sizeof(float), stream);   // bsum + bsq
        gcn_bn_stats<<<512, 256, 0, stream>>>(agg, norm_dst, Bs[l], bsum, bsq, N_NODES);
        gcn_bn_finalize<<<1, HID_C, 0, stream>>>(bsum, bsq, Gs[l], Bt[l],
                                                 bnscale, bnshift, 1.0f / N_NODES);
    }

    // FC: out = relu(bn2(x)) @ wfc + bfc  (BN/ReLU fused into A-staging)
    gcn_wmma_gemm<<<gemm_blocks, 32 * (OUT_C / 16), 0, stream>>>(
        agg, HID_C, HID_C, wfc, OUT_C,
        nullptr, bnscale, bnshift, 1, bfc, out, N_NODES);
}